// LTSMemory_34677565948785
// MI455X (gfx1250) — compile-verified
//
#include <hip/hip_runtime.h>
#include <hip/hip_bf16.h>
#include <math.h>

#define B_   4
#define T_   1024
#define D_   512
#define CAP_ 32768

typedef __attribute__((ext_vector_type(2))) float v2f;
typedef __attribute__((ext_vector_type(8))) float v8f;

// ---------------------------------------------------------------------------
// V_WMMA_F32_16X16X4_F32 wrapper: D(16x16,f32) = A(16x4,f32) * B(4x16,f32) + C
// A/B are 2 VGPRs per lane (64 f32 / 32 lanes).
// ---------------------------------------------------------------------------
__device__ __forceinline__ v8f wmma4(v2f a, v2f b, v8f c) {
  return __builtin_amdgcn_wmma_f32_16x16x4_f32(
      /*neg_a=*/false, a, /*neg_b=*/false, b,
      /*c_mod=*/(short)0, c, /*reuse_a=*/false, /*reuse_b=*/false);
}

// Row-major 16x4 fragment load (serves as A for X-rows and as B for mem-rows,
// since B[k,n] = mem[(c0+n)*D + k0+k] has the identical per-lane pattern):
//  lane<16 : row = r0+lane,   k = k0+{0,1}
//  lane>=16: row = r0+lane-16,k = k0+{2,3}
__device__ __forceinline__ v2f load_frag(const float* __restrict__ p,
                                         int r0, int ld, int k0, int lane) {
  const float* q = p + (size_t)(r0 + (lane & 15)) * ld + k0 + ((lane >> 4) << 1);
  v2f f; f.x = q[0]; f.y = q[1];
  return f;
}

// ---------------------------------------------------------------------------
// Pass: per-row softmax stats (max m, sum-exp Z) of X @ M^T * scale.
// One workgroup (4 waves) per 16-row tile; waves split the CAP range and the
// per-wave online (m,Z) are merged through LDS.
// ---------------------------------------------------------------------------
__global__ void stats_kernel(const float* __restrict__ X, const float* __restrict__ M,
                             float* __restrict__ rowmax, float* __restrict__ rowsum,
                             float scale) {
  __shared__ float lm[4][16], lz[4][16];
  const int lane = threadIdx.x & 31;
  const int wave = threadIdx.x >> 5;
  const int r0 = blockIdx.x * 16;   // flat row over B*T (T multiple of 16)

  float m[8], z[8];
#pragma unroll
  for (int v = 0; v < 8; ++v) { m[v] = -3.0e38f; z[v] = 0.f; }

  for (int c0 = wave * 16; c0 < CAP_; c0 += 16 * 4) {
    v8f acc = {};
    for (int k0 = 0; k0 < D_; k0 += 4) {
      v2f a = load_frag(X, r0, D_, k0, lane);
      v2f b = load_frag(M, c0, D_, k0, lane);
      acc = wmma4(a, b, acc);
    }
#pragma unroll
    for (int v = 0; v < 8; ++v) {
      float s = acc[v] * scale;
      float tmax = s;
#pragma unroll
      for (int off = 1; off < 16; off <<= 1) tmax = fmaxf(tmax, __shfl_xor(tmax, off));
      float mnew = fmaxf(m[v], tmax);
      float p = __expf(s - mnew);
      float tsum = p;
#pragma unroll
      for (int off = 1; off < 16; off <<= 1) tsum += __shfl_xor(tsum, off);
      z[v] = z[v] * __expf(m[v] - mnew) + tsum;
      m[v] = mnew;
    }
  }
  // lane 0 of each half holds rows v (+8 for upper half)
  if (lane == 0)
    for (int v = 0; v < 8; ++v) { lm[wave][v] = m[v]; lz[wave][v] = z[v]; }
  if (lane == 16)
    for (int v = 0; v < 8; ++v) { lm[wave][8 + v] = m[v]; lz[wave][8 + v] = z[v]; }
  __syncthreads();
  if (threadIdx.x < 16) {
    int t = threadIdx.x;
    float Mf = lm[0][t];
    for (int w = 1; w < 4; ++w) Mf = fmaxf(Mf, lm[w][t]);
    float Zf = 0.f;
    for (int w = 0; w < 4; ++w) Zf += lz[w][t] * __expf(lm[w][t] - Mf);
    rowmax[r0 + t] = Mf;
    rowsum[r0 + t] = Zf;
  }
}

// ---------------------------------------------------------------------------
// usage[b,c] = sum_t exp(scores[b,t,c] - m[b,t]) / Z[b,t]
// One wave per 16-column strip of one batch; loops all T recomputing tiles.
// ---------------------------------------------------------------------------
__global__ void usage_kernel(const float* __restrict__ keys, const float* __restrict__ mem,
                             const float* __restrict__ rowmax, const float* __restrict__ rowsum,
                             float* __restrict__ usage) {
  const int lane = threadIdx.x & 31;
  const int wave = threadIdx.x >> 5;
  const int gw = blockIdx.x * 4 + wave;        // over B * CAP/16
  const int b  = gw / (CAP_ / 16);
  const int c0 = (gw % (CAP_ / 16)) * 16;
  const int h8 = (lane >> 4) * 8;

  float uacc = 0.f;
  for (int t0 = 0; t0 < T_; t0 += 16) {
    const int r0 = b * T_ + t0;
    v8f acc = {};
    for (int k0 = 0; k0 < D_; k0 += 4) {
      v2f a  = load_frag(keys, r0, D_, k0, lane);
      v2f bb = load_frag(mem,  c0, D_, k0, lane);
      acc = wmma4(a, bb, acc);
    }
#pragma unroll
    for (int v = 0; v < 8; ++v) {
      int row = r0 + h8 + v;
      uacc += __expf(acc[v] - rowmax[row]) / rowsum[row];
    }
  }
  uacc += __shfl_xor(uacc, 16);                // combine the two lane halves (same N)
  if (lane < 16) usage[(size_t)b * CAP_ + c0 + lane] = uacc;
}

// ---------------------------------------------------------------------------
// combined[b] = mean over T of importance[b,:,0]
// ---------------------------------------------------------------------------
__global__ void combined_kernel(const float* __restrict__ imp, float* __restrict__ combined) {
  __shared__ float red[256];
  const int b = blockIdx.x;
  float s = 0.f;
  for (int i = threadIdx.x; i < T_; i += blockDim.x) s += imp[b * T_ + i];
  red[threadIdx.x] = s;
  __syncthreads();
  for (int st = 128; st > 0; st >>= 1) {
    if ((int)threadIdx.x < st) red[threadIdx.x] += red[threadIdx.x + st];
    __syncthreads();
  }
  if (threadIdx.x == 0) combined[b] = red[0] / (float)T_;
}

// ---------------------------------------------------------------------------
// Pack (priority, index) into u64 for descending sort w/ ties -> lower index.
// ---------------------------------------------------------------------------
__global__ void init_sort_kernel(const float* __restrict__ usage,
                                 const float* __restrict__ combined,
                                 unsigned long long* __restrict__ sortbuf) {
  const int i = blockIdx.x * blockDim.x + threadIdx.x;
  if (i >= B_ * CAP_) return;
  const int b = i / CAP_, c = i % CAP_;
  float p = combined[b] + 0.1f * usage[i];
  unsigned u = __float_as_uint(p);
  u = (u & 0x80000000u) ? ~u : (u | 0x80000000u);   // order-preserving map
  sortbuf[i] = ((unsigned long long)u << 32) | (unsigned long long)(0xFFFFFFFFu - (unsigned)c);
}

__global__ void bitonic_kernel(unsigned long long* __restrict__ buf, int j, int k) {
  const int tid = blockIdx.x * blockDim.x + threadIdx.x;
  if (tid >= B_ * CAP_) return;
  const int i = tid % CAP_;
  const int base = (tid / CAP_) * CAP_;
  const int ixj = i ^ j;
  if (ixj > i) {
    unsigned long long a = buf[base + i], b = buf[base + ixj];
    const bool desc = ((i & k) == 0);
    if (desc ? (a < b) : (a > b)) { buf[base + i] = b; buf[base + ixj] = a; }
  }
}

__global__ void extract_kernel(const unsigned long long* __restrict__ buf,
                               int* __restrict__ indices) {
  const int i = blockIdx.x * blockDim.x + threadIdx.x;
  if (i >= B_ * T_) return;
  const int b = i / T_, j = i % T_;
  indices[i] = (int)(0xFFFFFFFFu - (unsigned)(buf[(size_t)b * CAP_ + j] & 0xFFFFFFFFull));
}

__global__ void decay_kernel(const float* __restrict__ mem, float* __restrict__ nm) {
  const size_t i = (size_t)blockIdx.x * blockDim.x + threadIdx.x;
  if (i >= (size_t)CAP_ * D_ / 4) return;
  float4 v = ((const float4*)mem)[i];
  v.x *= 0.99f; v.y *= 0.99f; v.z *= 0.99f; v.w *= 0.99f;
  ((float4*)nm)[i] = v;
}

__global__ void scatter_kernel(const float* __restrict__ values,
                               const int* __restrict__ indices,
                               float* __restrict__ nm) {
  const int row = blockIdx.x;                  // over B*T
  const int slot = indices[row];
  const float4 v = ((const float4*)(values + (size_t)row * D_))[threadIdx.x];
  float* dst = nm + (size_t)slot * D_ + threadIdx.x * 4;
  const float r = 0.01f / (float)B_;           // MEM_NEW_DATA_RATE / B (batch mean)
  atomicAdd(dst + 0, r * v.x);
  atomicAdd(dst + 1, r * v.y);
  atomicAdd(dst + 2, r * v.z);
  atomicAdd(dst + 3, r * v.w);
}

// ---------------------------------------------------------------------------
// out[b,q,:] = softmax(Q @ NM^T / sqrt(D)) @ NM, using precomputed (m,Z).
// Workgroup (4 waves) per 16-row q-tile; waves split K=512 to build the 16x16
// P tile cooperatively via LDS, then each wave owns a 16x128 output slab.
// ---------------------------------------------------------------------------
__global__ void attn_out_kernel(const float* __restrict__ Q, const float* __restrict__ NM,
                                const float* __restrict__ rowmax, const float* __restrict__ rowsum,
                                float* __restrict__ out) {
  __shared__ float Spart[4][256];
  __shared__ float P[256];
  const int lane = threadIdx.x & 31;
  const int wave = threadIdx.x >> 5;
  const int r0 = blockIdx.x * 16;
  const int dbase = wave * 128;
  const int h = lane >> 4;
  const int n = lane & 15;
  const float scale = 0.044194173824159216f;   // 1/sqrt(512)

  float mr[8], zr[8];
#pragma unroll
  for (int v = 0; v < 8; ++v) {
    int row = r0 + 8 * h + v;
    mr[v] = rowmax[row];
    zr[v] = rowsum[row];
  }

  v8f acc[8] = {};
  for (int c0 = 0; c0 < CAP_; c0 += 16) {
    // partial S over this wave's K range [dbase, dbase+128)
    v8f s = {};
    for (int k0 = dbase; k0 < dbase + 128; k0 += 4) {
      v2f a = load_frag(Q,  r0, D_, k0, lane);
      v2f b = load_frag(NM, c0, D_, k0, lane);
      s = wmma4(a, b, s);
    }
#pragma unroll
    for (int v = 0; v < 8; ++v) Spart[wave][v * 32 + lane] = s[v];
    __syncthreads();

    if (wave == 0) {
#pragma unroll
      for (int v = 0; v < 8; ++v) {
        float sf = Spart[0][v * 32 + lane] + Spart[1][v * 32 + lane] +
                   Spart[2][v * 32 + lane] + Spart[3][v * 32 + lane];
        P[(v + 8 * h) * 16 + n] = __expf(sf * scale - mr[v]) / zr[v];
      }
    }
    __syncthreads();

    // acc[dc](16x16) += P(16x16) @ NM[c0:c0+16, dbase+dc*16 : +16]
#pragma unroll
    for (int kt = 0; kt < 4; ++kt) {
      v2f a;
      const int pi = n * 16 + kt * 4 + (h << 1);
      a.x = P[pi]; a.y = P[pi + 1];
#pragma unroll
      for (int dc = 0; dc < 8; ++dc) {
        const float* q = NM + (size_t)(c0 + kt * 4 + (h << 1)) * D_ + dbase + dc * 16 + n;
        v2f b; b.x = q[0]; b.y = q[D_];
        acc[dc] = wmma4(a, b, acc[dc]);
      }
    }
    __syncthreads();                           // protect Spart/P for next tile
  }

#pragma unroll
  for (int dc = 0; dc < 8; ++dc)
#pragma unroll
    for (int v = 0; v < 8; ++v)
      out[(size_t)(r0 + 8 * h + v) * D_ + dbase + dc * 16 + n] = acc[dc][v];
}

// ---------------------------------------------------------------------------
extern "C" void kernel_launch(void* const* d_in, const int* in_sizes, int n_in,
                              void* d_out, int out_size, void* d_ws, size_t ws_size,
                              hipStream_t stream) {
  const float* keys       = (const float*)d_in[0];
  const float* values     = (const float*)d_in[1];
  const float* importance = (const float*)d_in[2];
  const float* query      = (const float*)d_in[3];
  const float* mem        = (const float*)d_in[4];
  float* out = (float*)d_out;

  // workspace carve-up (256B aligned)
  char* p = (char*)d_ws;
  auto carve = [&](size_t bytes) { char* r = p; p += (bytes + 255) & ~(size_t)255; return r; };
  float* rowmax_s = (float*)carve(sizeof(float) * B_ * T_);
  float* rowsum_s = (float*)carve(sizeof(float) * B_ * T_);
  float* usage    = (float*)carve(sizeof(float) * B_ * CAP_);
  float* combined = (float*)carve(sizeof(float) * B_);
  unsigned long long* sortbuf = (unsigned long long*)carve(sizeof(unsigned long long) * B_ * CAP_);
  int*   indices  = (int*)carve(sizeof(int) * B_ * T_);
  float* rowmax_a = (float*)carve(sizeof(float) * B_ * T_);
  float* rowsum_a = (float*)carve(sizeof(float) * B_ * T_);
  float* new_mem  = (float*)carve(sizeof(float) * (size_t)CAP_ * D_);

  // 1) scores softmax stats (scale = 1)
  stats_kernel<<<(B_ * T_) / 16, 128, 0, stream>>>(keys, mem, rowmax_s, rowsum_s, 1.0f);
  // 2) usage accumulation
  usage_kernel<<<(B_ * CAP_ / 16) / 4, 128, 0, stream>>>(keys, mem, rowmax_s, rowsum_s, usage);
  // 3) importance mean
  combined_kernel<<<B_, 256, 0, stream>>>(importance, combined);
  // 4) pack sort keys
  init_sort_kernel<<<(B_ * CAP_ + 255) / 256, 256, 0, stream>>>(usage, combined, sortbuf);
  // 5) global bitonic sort (descending, ties -> lower index first)
  for (int k = 2; k <= CAP_; k <<= 1)
    for (int j = k >> 1; j > 0; j >>= 1)
      bitonic_kernel<<<(B_ * CAP_ + 255) / 256, 256, 0, stream>>>(sortbuf, j, k);
  // 6) top-1024 indices in rank order
  extract_kernel<<<(B_ * T_ + 255) / 256, 256, 0, stream>>>(sortbuf, indices);
  // 7) new_mem = 0.99 * mem
  decay_kernel<<<((size_t)CAP_ * D_ / 4 + 255) / 256, 256, 0, stream>>>(mem, new_mem);
  // 8) scatter-add 0.0025 * values into selected slots
  scatter_kernel<<<B_ * T_, 128, 0, stream>>>(values, indices, new_mem);
  // 9) attention softmax stats (scale = 1/sqrt(D))
  stats_kernel<<<(B_ * T_) / 16, 128, 0, stream>>>(query, new_mem, rowmax_a, rowsum_a,
                                                   0.044194173824159216f);
  // 10) attention output
  attn_out_kernel<<<(B_ * T_) / 16, 128, 0, stream>>>(query, new_mem, rowmax_a, rowsum_a, out);
}